// TPUMoELayer_40819369181852
// MI455X (gfx1250) — compile-verified
//
#include <hip/hip_runtime.h>

// Problem dims (match reference: B=2,S=2048,H=2048,E=8,D=2048,K=2)
#define T_TOK 4096   // B*S
#define H_DIM 2048
#define E_NUM 8
#define D_DIM 2048

// GEMM tiling: block tile 128(M) x 64(N), K tiled by 32.
// 8 waves; wave w owns rows [16w,16w+16), all 64 cols via 4 wmma accumulators.
#define BM 128
#define BN 64
#define BK 32
#define KPAD (BK + 2)   // row stride 34 floats = 136B: keeps 8B align, spreads banks

typedef float v2f __attribute__((ext_vector_type(2)));
typedef float v8f __attribute__((ext_vector_type(8)));

// ---------------------------------------------------------------------------
// CDNA5 async global->LDS staging (guarded; falls back to sync copies)
// Builtin signatures (from clang diagnostics):
//   b128: (AS1 v4i*, AS3 v4i*, imm offset, imm cpol)
//   b32 : (AS1 int*, AS3 int*, imm offset, imm cpol)
// ---------------------------------------------------------------------------
#if __has_builtin(__builtin_amdgcn_global_load_async_to_lds_b128) && \
    __has_builtin(__builtin_amdgcn_global_load_async_to_lds_b32) &&  \
    __has_builtin(__builtin_amdgcn_s_wait_asynccnt)
#define USE_ASYNC_LDS 1
#else
#define USE_ASYNC_LDS 0
#endif

typedef int v4i_t __attribute__((vector_size(16)));
typedef __attribute__((address_space(1))) v4i_t* g_v4i_ptr;
typedef __attribute__((address_space(3))) v4i_t* l_v4i_ptr;
typedef __attribute__((address_space(1))) int* g_i32_ptr;
typedef __attribute__((address_space(3))) int* l_i32_ptr;

__device__ __forceinline__ void stage_b128(const float* g, float* l) {
#if USE_ASYNC_LDS
  __builtin_amdgcn_global_load_async_to_lds_b128((g_v4i_ptr)g, (l_v4i_ptr)l, 0, 0);
#else
  *(float4*)l = *(const float4*)g;
#endif
}

__device__ __forceinline__ void stage_b32(const float* g, float* l) {
#if USE_ASYNC_LDS
  __builtin_amdgcn_global_load_async_to_lds_b32((g_i32_ptr)g, (l_i32_ptr)l, 0, 0);
#else
  *l = *g;
#endif
}

template <int N>
__device__ __forceinline__ void wait_async() {
#if USE_ASYNC_LDS
  __builtin_amdgcn_s_wait_asynccnt(N);
#endif
}

__device__ __forceinline__ v8f wmma4(v2f a, v2f b, v8f c) {
  // fp32 WMMA: D = A(16x4) * B(4x16) + C(16x16)
  return __builtin_amdgcn_wmma_f32_16x16x4_f32(false, a, false, b, (short)0, c,
                                               false, false);
}

__device__ __forceinline__ float gelu_tanh(float x) {
  // jax.nn.gelu default (approximate=True, tanh form)
  float x3 = x * x * x;
  float t = tanhf(0.7978845608028654f * (x + 0.044715f * x3));
  return 0.5f * x * (1.0f + t);
}

// Both operand tiles are stored k-minor in LDS, so every fragment is one
// aligned ds_load_b64 (lane holds K = kk+2*half, kk+2*half+1).
__device__ __forceinline__ void mma_tile(float (*As)[KPAD], float (*Bs)[KPAD],
                                         int mrow0, int half, int l16, v8f c[4]) {
#pragma unroll
  for (int kk = 0; kk < BK; kk += 4) {
    v2f a = *(const v2f*)&As[mrow0 + l16][kk + 2 * half];
    v2f b0 = *(const v2f*)&Bs[0 + l16][kk + 2 * half];
    v2f b1 = *(const v2f*)&Bs[16 + l16][kk + 2 * half];
    v2f b2 = *(const v2f*)&Bs[32 + l16][kk + 2 * half];
    v2f b3 = *(const v2f*)&Bs[48 + l16][kk + 2 * half];
    c[0] = wmma4(a, b0, c[0]);
    c[1] = wmma4(a, b1, c[1]);
    c[2] = wmma4(a, b2, c[2]);
    c[3] = wmma4(a, b3, c[3]);
  }
}

// ---------------------------------------------------------------------------
// Kernel 0: zero the expert counters and the combined accumulator
// ---------------------------------------------------------------------------
__global__ __launch_bounds__(256) void init_ws(int* __restrict__ counts,
                                               float* __restrict__ combined,
                                               size_t n_floats) {
  if (blockIdx.x == 0 && threadIdx.x < E_NUM) counts[threadIdx.x] = 0;
  size_t i = (size_t)blockIdx.x * blockDim.x + threadIdx.x;
  size_t stride = (size_t)gridDim.x * blockDim.x;
  float4* c4 = (float4*)combined;
  size_t n4 = n_floats >> 2;
  for (size_t j = i; j < n4; j += stride) c4[j] = make_float4(0.f, 0.f, 0.f, 0.f);
}

// ---------------------------------------------------------------------------
// Kernel 1: router (logits -> softmax -> top2 -> renorm) + per-expert gather
// One wave32 per token; 8 tokens per 256-thread block.
// ---------------------------------------------------------------------------
__global__ __launch_bounds__(256) void moe_router(
    const float* __restrict__ x, const float* __restrict__ rw,
    const float* __restrict__ rb, int* __restrict__ counts,
    int* __restrict__ tok_list, float* __restrict__ gate_list) {
  const int token = blockIdx.x * 8 + (threadIdx.x >> 5);
  const int lane = threadIdx.x & 31;

  float acc[E_NUM];
#pragma unroll
  for (int e = 0; e < E_NUM; ++e) acc[e] = 0.f;

  const float* xr = x + (size_t)token * H_DIM;
  for (int h = lane; h < H_DIM; h += 32) {
    const float xv = xr[h];
    const float* wr = rw + (size_t)h * E_NUM;  // router_w is [H,E] row-major
#pragma unroll
    for (int e = 0; e < E_NUM; ++e) acc[e] += xv * wr[e];
  }
#pragma unroll
  for (int off = 16; off > 0; off >>= 1) {
#pragma unroll
    for (int e = 0; e < E_NUM; ++e) acc[e] += __shfl_xor(acc[e], off, 32);
  }

  if (lane == 0) {
    float l[E_NUM], p[E_NUM];
    float m = -1e30f;
#pragma unroll
    for (int e = 0; e < E_NUM; ++e) {
      l[e] = acc[e] + rb[e];
      m = fmaxf(m, l[e]);
    }
    float s = 0.f;
#pragma unroll
    for (int e = 0; e < E_NUM; ++e) {
      p[e] = __expf(l[e] - m);
      s += p[e];
    }
    float inv = 1.f / s;
#pragma unroll
    for (int e = 0; e < E_NUM; ++e) p[e] *= inv;

    int i1 = 0;
    float p1 = p[0];
#pragma unroll
    for (int e = 1; e < E_NUM; ++e)
      if (p[e] > p1) { p1 = p[e]; i1 = e; }
    int i2 = (i1 == 0) ? 1 : 0;
    float p2 = p[i2];
#pragma unroll
    for (int e = 0; e < E_NUM; ++e)
      if (e != i1 && p[e] > p2) { p2 = p[e]; i2 = e; }

    const float rs = 1.f / (p1 + p2);
    int s1 = atomicAdd(&counts[i1], 1);
    tok_list[i1 * T_TOK + s1] = token;
    gate_list[i1 * T_TOK + s1] = p1 * rs;
    int s2 = atomicAdd(&counts[i2], 1);
    tok_list[i2 * T_TOK + s2] = token;
    gate_list[i2 * T_TOK + s2] = p2 * rs;
  }
}

// ---------------------------------------------------------------------------
// Kernel 2: grouped expert GEMM over gathered tokens (top-2 only).
// Double-buffered async global->LDS staging; wave = 16x64 via 4 WMMA accums.
// Epilogue: +bias, gelu, *gate, atomic accumulate into combined[token, d].
// ---------------------------------------------------------------------------
__global__ __launch_bounds__(256) void moe_expert_gemm(
    const float* __restrict__ x, const float* __restrict__ ew,
    const float* __restrict__ eb, const int* __restrict__ counts,
    const int* __restrict__ tok_list, const float* __restrict__ gate_list,
    float* __restrict__ combined) {
  const int e = blockIdx.y;
  const int cnt = counts[e];
  const int ntiles = D_DIM / BN;  // 32
  const int mtile = blockIdx.x / ntiles;
  const int ntile = blockIdx.x % ntiles;
  const int mbase = mtile * BM;
  if (mbase >= cnt) return;  // over-provisioned grid: idle tiles exit fast
  const int mrem = cnt - mbase;
  const int nbase = ntile * BN;

  __shared__ float As[2][BM][KPAD];  // token rows, k-minor
  __shared__ float Bs[2][BN][KPAD];  // weight tile TRANSPOSED (n-major, k-minor)
  __shared__ int toks[BM];
  __shared__ float gts[BM];

  const int tid = threadIdx.x;
  if (tid < BM) {
    if (tid < mrem) {
      toks[tid] = tok_list[e * T_TOK + mbase + tid];
      gts[tid] = gate_list[e * T_TOK + mbase + tid];
    } else {
      toks[tid] = 0;  // padded rows compute garbage; stores are guarded
      gts[tid] = 0.f;
    }
  }
  __syncthreads();

  const float* wbase = ew + (size_t)e * H_DIM * D_DIM;

  // 12 async ops per thread per tile: A = 4 x b128, B = 8 x b32 (transposing)
  auto stage = [&](int buf, int k0) {
#pragma unroll
    for (int rep = 0; rep < 4; ++rep) {
      int idx = tid + rep * 256;
      int row = idx >> 3;          // 8 float4 per 32-float row
      int col = (idx & 7) * 4;
      stage_b128(x + (size_t)toks[row] * H_DIM + k0 + col, &As[buf][row][col]);
    }
#pragma unroll
    for (int rep = 0; rep < 8; ++rep) {
      int idx = tid + rep * 256;   // 2048 elements, lane-consecutive n
      int k = idx >> 6;
      int n = idx & 63;
      stage_b32(wbase + (size_t)(k0 + k) * D_DIM + nbase + n, &Bs[buf][n][k]);
    }
  };

  const int wave = tid >> 5;
  const int lane = tid & 31;
  const int half = lane >> 4;
  const int l16 = lane & 15;
  const int mrow0 = wave * 16;

  v8f c[4] = {{}, {}, {}, {}};

  const int NT = H_DIM / BK;  // 64
  stage(0, 0);
  for (int t = 0; t < NT - 1; ++t) {
    stage((t + 1) & 1, (t + 1) * BK);
    wait_async<12>();  // tile t's 12 ops (oldest) complete; next 12 in flight
    __syncthreads();
    mma_tile(As[t & 1], Bs[t & 1], mrow0, half, l16, c);
    __syncthreads();
  }
  wait_async<0>();
  __syncthreads();
  mma_tile(As[(NT - 1) & 1], Bs[(NT - 1) & 1], mrow0, half, l16, c);

  // Epilogue: C/D layout -> VGPR r holds (M = mrow0 + 8*half + r, N = 16*i + l16)
#pragma unroll
  for (int i = 0; i < 4; ++i) {
    const int gcol = nbase + i * 16 + l16;
    const float bias = eb[e * D_DIM + gcol];
#pragma unroll
    for (int r = 0; r < 8; ++r) {
      int row = mrow0 + half * 8 + r;
      if (row < mrem) {
        atomicAdd(&combined[(size_t)toks[row] * D_DIM + gcol],
                  gelu_tanh(c[i][r] + bias) * gts[row]);
      }
    }
  }
}

// ---------------------------------------------------------------------------
// Kernel 3: output projection  out = combined @ out_w + out_b
// ---------------------------------------------------------------------------
__global__ __launch_bounds__(256) void moe_out_proj(
    const float* __restrict__ combined, const float* __restrict__ ow,
    const float* __restrict__ ob, float* __restrict__ out) {
  const int ntiles = D_DIM / BN;
  const int mtile = blockIdx.x / ntiles;
  const int ntile = blockIdx.x % ntiles;
  const int nbase = ntile * BN;

  __shared__ float As[2][BM][KPAD];
  __shared__ float Bs[2][BN][KPAD];

  const int tid = threadIdx.x;

  auto stage = [&](int buf, int k0) {
#pragma unroll
    for (int rep = 0; rep < 4; ++rep) {
      int idx = tid + rep * 256;
      int row = idx >> 3;
      int col = (idx & 7) * 4;
      stage_b128(combined + (size_t)(mtile * BM + row) * D_DIM + k0 + col,
                 &As[buf][row][col]);
    }
#pragma unroll
    for (int rep = 0; rep < 8; ++rep) {
      int idx = tid + rep * 256;
      int k = idx >> 6;
      int n = idx & 63;
      stage_b32(ow + (size_t)(k0 + k) * D_DIM + nbase + n, &Bs[buf][n][k]);
    }
  };

  const int wave = tid >> 5;
  const int lane = tid & 31;
  const int half = lane >> 4;
  const int l16 = lane & 15;
  const int mrow0 = wave * 16;

  v8f c[4] = {{}, {}, {}, {}};

  const int NT = D_DIM / BK;  // 64
  stage(0, 0);
  for (int t = 0; t < NT - 1; ++t) {
    stage((t + 1) & 1, (t + 1) * BK);
    wait_async<12>();
    __syncthreads();
    mma_tile(As[t & 1], Bs[t & 1], mrow0, half, l16, c);
    __syncthreads();
  }
  wait_async<0>();
  __syncthreads();
  mma_tile(As[(NT - 1) & 1], Bs[(NT - 1) & 1], mrow0, half, l16, c);

#pragma unroll
  for (int i = 0; i < 4; ++i) {
    const int gcol = nbase + i * 16 + l16;
    const float bias = ob[gcol];
#pragma unroll
    for (int r = 0; r < 8; ++r) {
      int grow = mtile * BM + mrow0 + half * 8 + r;
      out[(size_t)grow * D_DIM + gcol] = c[i][r] + bias;
    }
  }
}

// ---------------------------------------------------------------------------
extern "C" void kernel_launch(void* const* d_in, const int* in_sizes, int n_in,
                              void* d_out, int out_size, void* d_ws,
                              size_t ws_size, hipStream_t stream) {
  (void)in_sizes; (void)n_in; (void)out_size; (void)ws_size;
  const float* x  = (const float*)d_in[0];  // [B,S,H]
  const float* rw = (const float*)d_in[1];  // [H,E]
  const float* rb = (const float*)d_in[2];  // [E]
  const float* ew = (const float*)d_in[3];  // [E,H,D]
  const float* eb = (const float*)d_in[4];  // [E,D]
  const float* ow = (const float*)d_in[5];  // [D,D]
  const float* ob = (const float*)d_in[6];  // [D]
  float* out = (float*)d_out;               // [B,S,D]

  // Workspace layout
  char* ws = (char*)d_ws;
  int* counts      = (int*)(ws);                                  // E ints
  int* tok_list    = (int*)(ws + 1024);                           // E*T ints
  float* gate_list = (float*)(ws + 1024 + E_NUM * T_TOK * 4);     // E*T floats
  float* combined  = (float*)(ws + 1024 + 2 * E_NUM * T_TOK * 4); // T*D floats

  init_ws<<<2048, 256, 0, stream>>>(counts, combined, (size_t)T_TOK * D_DIM);
  moe_router<<<T_TOK / 8, 256, 0, stream>>>(x, rw, rb, counts, tok_list,
                                            gate_list);
  dim3 eg((T_TOK / BM) * (D_DIM / BN), E_NUM);  // worst case: all tokens -> 1 expert
  moe_expert_gemm<<<eg, 256, 0, stream>>>(x, ew, eb, counts, tok_list,
                                          gate_list, combined);
  moe_out_proj<<<(T_TOK / BM) * (D_DIM / BN), 256, 0, stream>>>(combined, ow,
                                                                ob, out);
}